// ROIPooling1_5746666242434
// MI455X (gfx1250) — compile-verified
//
#include <hip/hip_runtime.h>

// ROI max-pool, MI455X (gfx1250, wave32).
// Memory-bound: ~75MB reads + 6.4MB writes; the whole feature map (8MB) is
// L2-resident (192MB L2) -> kernel is issue/latency bound, not HBM bound.
// No matmul structure -> WMMA inapplicable. CDNA5-specific path: async
// global->LDS DMA (GLOBAL_LOAD_ASYNC_TO_LDS_B128, ASYNCcnt) stages each
// channel's ROI bounding box into LDS; 49 lanes then do the bin max from LDS.

#define B_ 2
#define C_ 256
#define H_ 64
#define W_ 64
#define R_ 64
#define P_ 7
#define NEGF (-3e38f)

// b32 builtin params: AS(1) int* / AS(3) int* (round-1 diagnostic).
// b128 builtin params: AS(1) v4i* / AS(3) v4i* (round-3 diagnostic:
// '__attribute__((__vector_size__(4 * sizeof(int)))) int __device__ *').
typedef __attribute__((vector_size(16))) int int4v;
typedef __attribute__((address_space(1))) int   gint_as1;
typedef __attribute__((address_space(3))) int   lint_as3;
typedef __attribute__((address_space(1))) int4v gv4i_as1;
typedef __attribute__((address_space(3))) int4v lv4i_as3;

__device__ __forceinline__ void wait_async_done() {
#if __has_builtin(__builtin_amdgcn_s_wait_asynccnt)
  __builtin_amdgcn_s_wait_asynccnt(0);
#elif defined(__AMDGCN__)
  asm volatile("s_wait_asynccnt 0" ::: "memory");
#endif
}

__global__ __launch_bounds__(64)
void roi_pool_gfx1250(const float* __restrict__ feat,
                      const float* __restrict__ rois,
                      float* __restrict__ out) {
  // Worst case: 64 rows x (64-wide row + 4 pad) floats = 17.4KB.
  __shared__ float smem[64 * 68];

  const int c   = blockIdx.x;   // channel fastest: adjacent blocks share ROI rows in L2
  const int r   = blockIdx.y;
  const int b   = blockIdx.z;
  const int tid = threadIdx.x;

  // ---- ROI geometry (block-uniform -> scalar float/int ops) -----------------
  const float* roi = rois + (size_t)(b * R_ + r) * 4;
  const int px = __float2int_rn(roi[0] * 0.0625f);
  const int py = __float2int_rn(roi[1] * 0.0625f);
  const int qx = __float2int_rn(roi[2] * 0.0625f);
  const int qy = __float2int_rn(roi[3] * 0.0625f);

  int roi_w = qx - px + 1; if (roi_w < 1) roi_w = 1;
  int roi_h = qy - py + 1; if (roi_h < 1) roi_h = 1;
  const int psw   = (roi_w + P_ - 1) / P_;
  const int psh   = (roi_h + P_ - 1) / P_;
  const int pad_l = (psw * P_ - roi_w) >> 1;
  const int pad_t = (psh * P_ - roi_h) >> 1;

  // Widen the staged span to 16B-aligned bounds; stays inside the 64-wide row.
  const int x0a = px & ~3;                       // aligned left edge
  int x1a = (px + roi_w + 3) & ~3;               // aligned right edge (<= 64)
  if (x1a > W_) x1a = W_;
  const int wr    = x1a - x0a;                   // widened width, multiple of 4
  const int w4    = wr >> 2;                     // quads per row (<= 16)
  const int ldstr = wr + 4;                      // LDS stride: 16B-aligned rows + pad
  const int off   = px - x0a;                    // bin-phase x offset (0..3)

  // ---- Stage ROI bounding box of this channel into LDS (async DMA) ----------
  const float* fbase = feat + (size_t)(b * C_ + c) * (H_ * W_);
#if __has_builtin(__builtin_amdgcn_global_load_async_to_lds_b128)
  {
    // 64 lanes = 4 rows x 16 quads: every b128 DMA moves 16B, <=16 issues/lane.
    const int ry = tid >> 4;      // row group 0..3
    const int rx = tid & 15;      // quad index within row
    if (rx < w4) {
      for (int y = ry; y < roi_h; y += 4) {
        const float* src = fbase + (py + y) * W_ + x0a + rx * 4;  // 16B aligned
        float*       dst = smem + y * ldstr + rx * 4;             // 16B aligned
        __builtin_amdgcn_global_load_async_to_lds_b128(
            (gv4i_as1*)src, (lv4i_as3*)dst, /*imm offset*/0, /*cpol*/0);
      }
    }
  }
#elif __has_builtin(__builtin_amdgcn_global_load_async_to_lds_b32)
  for (int y = 0; y < roi_h; ++y) {
    const float* src = fbase + (py + y) * W_ + x0a;
    float*       dst = smem + y * ldstr;
    for (int x = tid; x < wr; x += 64)
      __builtin_amdgcn_global_load_async_to_lds_b32(
          (gint_as1*)(src + x), (lint_as3*)(dst + x), 0, 0);
  }
#else
  for (int y = 0; y < roi_h; ++y) {
    const float* src = fbase + (py + y) * W_ + x0a;
    float*       dst = smem + y * ldstr;
    for (int x = tid; x < wr; x += 64) dst[x] = src[x];
  }
#endif
  wait_async_done();   // each wave drains its own ASYNCcnt (LDS writes landed)
  __syncthreads();     // make all waves' staged data visible

  // ---- Per-bin max reduction from LDS ---------------------------------------
  if (tid < P_ * P_) {
    const int i = tid / P_;        // y-bin
    const int j = tid - i * P_;    // x-bin

    int y0 = i * psh - pad_t;  int y1 = y0 + psh;
    int x0 = j * psw - pad_l;  int x1 = x0 + psw;
    if (y0 < 0) y0 = 0;  if (y1 > roi_h) y1 = roi_h;
    if (x0 < 0) x0 = 0;  if (x1 > roi_w) x1 = roi_w;

    float m = NEGF;                         // empty window stays NEG (then pad->0)
    for (int y = y0; y < y1; ++y) {
      const float* row = smem + y * ldstr + off;
      for (int x = x0; x < x1; ++x)
        m = fmaxf(m, row[x]);
    }

    const bool padr = (i * psh < pad_t) || ((i + 1) * psh > pad_t + roi_h);
    const bool padc = (j * psw < pad_l) || ((j + 1) * psw > pad_l + roi_w);
    if (padr || padc) m = fmaxf(m, 0.0f);   // reference: where(padf, max(pooled,0), pooled)

    out[((size_t)(b * R_ + r) * C_ + c) * (P_ * P_) + tid] = m;
  }
}

extern "C" void kernel_launch(void* const* d_in, const int* in_sizes, int n_in,
                              void* d_out, int out_size, void* d_ws, size_t ws_size,
                              hipStream_t stream) {
  const float* feat = (const float*)d_in[0];  // (2,256,64,64) f32
  const float* rois = (const float*)d_in[1];  // (2,64,4) f32
  float* out = (float*)d_out;                 // (2,64,256,7,7) f32

  dim3 grid(C_, R_, B_);   // 256 x 64 x 2 = 32768 blocks
  roi_pool_gfx1250<<<grid, 64, 0, stream>>>(feat, rois, out);
  (void)in_sizes; (void)n_in; (void)out_size; (void)d_ws; (void)ws_size;
}